// TrxMeanEncoder_69681549410810
// MI455X (gfx1250) — compile-verified
//
#include <hip/hip_runtime.h>
#include <math.h>

// ---------------------------------------------------------------------------
// TrxMeanEncoder for MI455X (gfx1250, wave32)
//   out[b, 0:400]   = per-row counts of mcc_code / 512
//   out[b, 400:500] = per-row counts of tr_type  / 512
//   out[b, 500]     = sum_{t<seq_len} sign(a)*log1p(|a|) / seq_len
// Memory-bound (~129 MB total traffic -> ~5.6 us at 23.3 TB/s).
// log1p computed via native v_log_f32 (~1 ulp), keeping VALU far below the
// memory roofline (precise log1pf costs ~55 VALU/elt and would dominate).
// ---------------------------------------------------------------------------

typedef __attribute__((ext_vector_type(2))) float v2f;
typedef __attribute__((ext_vector_type(8))) float v8f;

#define T_LEN    512
#define MCC_DIM  400
#define TR_DIM   100
#define OUT_COLS 501

// ln(2): converts v_log_f32 (log2) to natural log
#define LN2_F 0.69314718055994530942f

// ---------------------------------------------------------------------------
// Kernel 1: per-row histogram in LDS (ds_add atomics), writes cols 0..499.
// One block (256 threads) per row; int2 loads are fully coalesced.
// ---------------------------------------------------------------------------
__global__ __launch_bounds__(256) void hist_kernel(
    const int* __restrict__ mcc, const int* __restrict__ tr,
    float* __restrict__ out)
{
    __shared__ unsigned hist[512];  // 0..399 mcc, 400..499 tr, rest padding

    const int row = blockIdx.x;
    const int tid = threadIdx.x;

    hist[tid]       = 0u;
    hist[tid + 256] = 0u;
    __syncthreads();

    const int2 mv = ((const int2*)(mcc + (size_t)row * T_LEN))[tid];
    const int2 tv = ((const int2*)(tr  + (size_t)row * T_LEN))[tid];

    atomicAdd(&hist[mv.x], 1u);
    atomicAdd(&hist[mv.y], 1u);
    atomicAdd(&hist[MCC_DIM + tv.x], 1u);
    atomicAdd(&hist[MCC_DIM + tv.y], 1u);
    __syncthreads();

    float* orow = out + (size_t)row * OUT_COLS;
    const float inv = 1.0f / (float)T_LEN;
    orow[tid] = (float)hist[tid] * inv;
    if (tid < (MCC_DIM + TR_DIM - 256))
        orow[tid + 256] = (float)hist[tid + 256] * inv;
}

// Branch-free sign(x)*log1p(|x|) via the native log2 transcendental.
__device__ __forceinline__ float signed_log1p_fast(float x)
{
    const float ax = fabsf(x);
    const float lg = __builtin_amdgcn_logf(1.0f + ax) * LN2_F;  // v_log_f32
    return copysignf(lg, x);                                    // v_bfi
}

// ---------------------------------------------------------------------------
// Kernel 2: masked row-sums via V_WMMA_F32_16X16X4_F32 (D = A x ones + C).
// Each wave: one 16-row tile x one K-half (256 cols) -> 2048 waves total.
// A layout (16x4 f32): lanes 0-15 hold M=0..15 {K=0,K=1}; lanes 16-31 {K=2,K=3}.
// D layout (16x16 f32): VGPR r -> M=r (lanes 0-15), M=8+r (lanes 16-31).
// Partials go to workspace: partial[row*2 + khalf].
// Inner loop is branch-free (cndmask select) so the float2 loads stay b64
// and EXEC remains all-ones for every WMMA.
// ---------------------------------------------------------------------------
__global__ __launch_bounds__(256) void mean_partial_kernel(
    const float* __restrict__ amount, const int* __restrict__ seq_lens,
    float* __restrict__ partial)
{
    const int lane  = threadIdx.x & 31;
    const int wave  = blockIdx.x * (blockDim.x >> 5) + (threadIdx.x >> 5);
    const int tile  = wave >> 1;      // 16-row tile index
    const int khalf = wave & 1;       // which half of the K range
    const int m     = lane & 15;      // row within tile
    const int half  = lane >> 4;      // A-matrix K-pair select

    const int row = tile * 16 + m;
    const float* __restrict__ arow = amount + (size_t)row * T_LEN;
    const int ilen = seq_lens[row];

    v2f bones;
    bones[0] = 1.0f;
    bones[1] = 1.0f;
    v8f acc = {};  // f32 accumulator, zero C

    const int kbeg = khalf * (T_LEN / 2);
    #pragma unroll 4
    for (int k0 = 0; k0 < T_LEN / 2; k0 += 4) {
        const int k = kbeg + k0 + 2 * half;
        const float2 x = *(const float2*)(arow + k);  // 8B-aligned b64 load

        // Unconditional cheap compute, then arithmetic mask (no branches).
        float v0 = signed_log1p_fast(x.x);
        float v1 = signed_log1p_fast(x.y);
        v0 = (k     < ilen) ? v0 : 0.0f;
        v1 = (k + 1 < ilen) ? v1 : 0.0f;

        v2f a;
        a[0] = v0;
        a[1] = v1;

        // D[m,n] = sum_k A[m,k]*1 + C[m,n]  (every column holds the row sum)
        acc = __builtin_amdgcn_wmma_f32_16x16x4_f32(
            /*neg_a=*/false, a, /*neg_b=*/false, bones,
            /*c_mod=*/(short)0, acc, /*reuse_a=*/false, /*reuse_b=*/false);
    }

    // Lane 0 holds rows tile*16+0..7 in acc[0..7]; lane 16 holds rows +8..15.
    if (lane == 0 || lane == 16) {
        const int rbase = tile * 16 + 8 * half;
        #pragma unroll
        for (int r = 0; r < 8; ++r)
            partial[(size_t)(rbase + r) * 2 + khalf] = acc[r];
    }
}

// ---------------------------------------------------------------------------
// Kernel 3: combine K-half partials, divide by seq_len, write col 500.
// ---------------------------------------------------------------------------
__global__ __launch_bounds__(256) void mean_final_kernel(
    const float* __restrict__ partial, const int* __restrict__ seq_lens,
    float* __restrict__ out, int B)
{
    const int row = blockIdx.x * blockDim.x + threadIdx.x;
    if (row >= B) return;
    const float s = partial[(size_t)row * 2] + partial[(size_t)row * 2 + 1];
    out[(size_t)row * OUT_COLS + 500] = s / (float)seq_lens[row];
}

// ---------------------------------------------------------------------------
extern "C" void kernel_launch(void* const* d_in, const int* in_sizes, int n_in,
                              void* d_out, int out_size, void* d_ws, size_t ws_size,
                              hipStream_t stream)
{
    const int*   mcc    = (const int*)d_in[0];
    const int*   tr     = (const int*)d_in[1];
    const float* amount = (const float*)d_in[2];
    const int*   seq    = (const int*)d_in[3];
    float*       out    = (float*)d_out;

    const int B = in_sizes[3];          // 16384 rows (multiple of 64)
    float* partial = (float*)d_ws;      // B*2 floats = 128 KB scratch

    // Cols 0..499: LDS histogram, one block per row.
    hist_kernel<<<B, 256, 0, stream>>>(mcc, tr, out);

    // Col 500: WMMA masked row-sums (2 K-halves per 16-row tile).
    const int waves2  = (B / 16) * 2;   // 2048
    const int blocks2 = waves2 / 8;     // 8 waves (256 thr) per block
    mean_partial_kernel<<<blocks2, 256, 0, stream>>>(amount, seq, partial);

    mean_final_kernel<<<(B + 255) / 256, 256, 0, stream>>>(partial, seq, out, B);
}